// GCNLayers_21835613733678
// MI455X (gfx1250) — compile-verified
//
#include <hip/hip_runtime.h>
#include <hip/hip_bf16.h>

typedef float v2f __attribute__((ext_vector_type(2)));
typedef float v8f __attribute__((ext_vector_type(8)));

// ---------------------------------------------------------------------------
// Degree kernels: deg[j] = 1 (self loop) + #edges with dst==j ; then rsqrt.
// ---------------------------------------------------------------------------
__global__ void k_deg_init(float* __restrict__ deg, int n) {
  int i = blockIdx.x * blockDim.x + threadIdx.x;
  if (i < n) deg[i] = 1.0f;
}

__global__ void k_deg_accum(const int* __restrict__ dst, float* __restrict__ deg, int e) {
  int i = blockIdx.x * blockDim.x + threadIdx.x;
  if (i < e) atomicAdd(&deg[dst[i]], 1.0f);
}

__global__ void k_deg_rsqrt(float* __restrict__ deg, int n) {
  int i = blockIdx.x * blockDim.x + threadIdx.x;
  if (i < n) deg[i] = rsqrtf(deg[i]);  // deg >= 1 always (self loops)
}

// ---------------------------------------------------------------------------
// FP32 WMMA GEMM: C[M,N] = A[M,K] * B[K,N], via V_WMMA_F32_16X16X4_F32.
// Each wave computes a 32(M) x 32(N) tile with 4 accumulators so A/B loads
// are amortized: per k-step 2 x b64 (A) + 4 x b32 (B) feed 4 WMMAs.
// Block = one 32-row strip; wave w covers N columns [w*32, w*32+32).
// Requires K % 4 == 0, N % 32 == 0; M guarded (loads clamped, stores masked).
// Layout (ISA 7.12.2, 32-bit operands), lane l: m = l&15, h = l>>4:
//   A reg r : A[m][k0 + 2h + r]      B reg r : B[k0 + 2h + r][n]
//   C reg r : row (r + 8h), col n
// ---------------------------------------------------------------------------
__global__ void k_gemm_wmma_f32(const float* __restrict__ A,
                                const float* __restrict__ B,
                                float* __restrict__ C,
                                int M, int K, int N) {
  const int lane = threadIdx.x & 31;
  const int wv   = threadIdx.x >> 5;   // n-group within block
  const int h    = lane >> 4;
  const int l16  = lane & 15;

  const int mbase = blockIdx.x * 32;
  const int m0 = mbase + l16;          // rows for A tile 0
  const int m1 = m0 + 16;              // rows for A tile 1
  const int mc0 = (m0 < M) ? m0 : (M - 1);
  const int mc1 = (m1 < M) ? m1 : (M - 1);
  const int n0 = wv * 32 + l16;        // cols for B tile 0
  const int n1 = n0 + 16;              // cols for B tile 1

  const float* a0row = A + (size_t)mc0 * K;
  const float* a1row = A + (size_t)mc1 * K;

  v8f acc00 = {}, acc01 = {}, acc10 = {}, acc11 = {};
  for (int k0 = 0; k0 < K; k0 += 4) {
    const int ka = k0 + 2 * h;
    v2f a0, a1, b0, b1;
    a0.x = a0row[ka];  a0.y = a0row[ka + 1];
    a1.x = a1row[ka];  a1.y = a1row[ka + 1];
    const float* br0 = B + (size_t)ka * N;
    const float* br1 = br0 + N;
    b0.x = br0[n0];  b0.y = br1[n0];
    b1.x = br0[n1];  b1.y = br1[n1];
    acc00 = __builtin_amdgcn_wmma_f32_16x16x4_f32(false, a0, false, b0, (short)0, acc00, false, false);
    acc01 = __builtin_amdgcn_wmma_f32_16x16x4_f32(false, a0, false, b1, (short)0, acc01, false, false);
    acc10 = __builtin_amdgcn_wmma_f32_16x16x4_f32(false, a1, false, b0, (short)0, acc10, false, false);
    acc11 = __builtin_amdgcn_wmma_f32_16x16x4_f32(false, a1, false, b1, (short)0, acc11, false, false);
  }

#pragma unroll
  for (int r = 0; r < 8; ++r) {
    const int row0 = mbase + r + 8 * h;
    const int row1 = row0 + 16;
    if (row0 < M) {
      C[(size_t)row0 * N + n0] = acc00[r];
      C[(size_t)row0 * N + n1] = acc01[r];
    }
    if (row1 < M) {
      C[(size_t)row1 * N + n0] = acc10[r];
      C[(size_t)row1 * N + n1] = acc11[r];
    }
  }
}

// ---------------------------------------------------------------------------
// out[i, c4..c4+3] = H[i, c4..c4+3] * dinv[i]^2  [+ bias]   (float4 lanes)
// ---------------------------------------------------------------------------
__global__ void k_selfloop_init_v4(const float* __restrict__ H,
                                   const float* __restrict__ dinv,
                                   const float* __restrict__ bias,
                                   float* __restrict__ out,
                                   int n, int C, int add_bias) {
  int idx = blockIdx.x * blockDim.x + threadIdx.x;      // float4 index
  int c4 = C >> 2;
  if (idx >= n * c4) return;
  int i = idx / c4;
  int c = (idx - i * c4) << 2;
  float w = dinv[i];
  w *= w;
  float4 hv = reinterpret_cast<const float4*>(H)[idx];
  float4 v = make_float4(hv.x * w, hv.y * w, hv.z * w, hv.w * w);
  if (add_bias) {
    float4 bv = *reinterpret_cast<const float4*>(bias + c);
    v.x += bv.x; v.y += bv.y; v.z += bv.z; v.w += bv.w;
  }
  reinterpret_cast<float4*>(out)[idx] = v;
}

// ---------------------------------------------------------------------------
// Edge scatter: out[dst] += dinv[src]*dinv[dst] * H[src].
// One wave per edge; each lane handles VEC contiguous floats so the gather is
// one wide vector load (b128 for C=128, b64 for C=64) + VEC f32 atomics.
// H and out are L2-resident (<< 192MB), atomics resolve at L2.
// Requires C == 32*VEC.
// ---------------------------------------------------------------------------
template <int VEC>
__global__ void k_edge_scatter(const int* __restrict__ src,
                               const int* __restrict__ dst,
                               const float* __restrict__ H,
                               const float* __restrict__ dinv,
                               float* __restrict__ out,
                               int E, int C) {
  int lane = threadIdx.x & 31;
  int e = (blockIdx.x * blockDim.x + threadIdx.x) >> 5;
  if (e >= E) return;
  int i = src[e];
  int j = dst[e];
  float w = dinv[i] * dinv[j];
  const float* hrow = H + (size_t)i * C + lane * VEC;
  float* orow = out + (size_t)j * C + lane * VEC;
  if constexpr (VEC == 4) {
    float4 hv = *reinterpret_cast<const float4*>(hrow);
    atomicAdd(orow + 0, hv.x * w);
    atomicAdd(orow + 1, hv.y * w);
    atomicAdd(orow + 2, hv.z * w);
    atomicAdd(orow + 3, hv.w * w);
  } else {
    float2 hv = *reinterpret_cast<const float2*>(hrow);
    atomicAdd(orow + 0, hv.x * w);
    atomicAdd(orow + 1, hv.y * w);
  }
}

// ---------------------------------------------------------------------------
// a[i,c] = relu(a[i,c] + b[c])   (float4 lanes)
// ---------------------------------------------------------------------------
__global__ void k_bias_relu_v4(float* __restrict__ a, const float* __restrict__ b,
                               int n, int C) {
  int idx = blockIdx.x * blockDim.x + threadIdx.x;
  int c4 = C >> 2;
  if (idx >= n * c4) return;
  int i = idx / c4;
  int c = (idx - i * c4) << 2;
  float4 v = reinterpret_cast<float4*>(a)[idx];
  float4 bv = *reinterpret_cast<const float4*>(b + c);
  v.x = fmaxf(v.x + bv.x, 0.0f);
  v.y = fmaxf(v.y + bv.y, 0.0f);
  v.z = fmaxf(v.z + bv.z, 0.0f);
  v.w = fmaxf(v.w + bv.w, 0.0f);
  reinterpret_cast<float4*>(a)[idx] = v;
}

extern "C" void kernel_launch(void* const* d_in, const int* in_sizes, int n_in,
                              void* d_out, int out_size, void* d_ws, size_t ws_size,
                              hipStream_t stream) {
  const int CIN = 256, CH = 128, CO = 64;
  const float* x  = (const float*)d_in[0];
  const int*   ei = (const int*)d_in[1];
  const float* W1 = (const float*)d_in[2];
  const float* b1 = (const float*)d_in[3];
  const float* W2 = (const float*)d_in[4];
  const float* b2 = (const float*)d_in[5];
  float* out = (float*)d_out;

  const int N = in_sizes[0] / CIN;   // 50000
  const int E = in_sizes[1] / 2;     // 800000
  const int* src = ei;
  const int* dst = ei + E;

  // Workspace layout (floats): [dinv: N][H: N*CH (reused for N*CO)][AGG: N*CH]
  float* ws   = (float*)d_ws;
  float* dinv = ws;
  float* Hbuf = ws + 50176;                 // aligned past dinv
  float* AGG  = ws + 50176 + (size_t)N * CH;

  const int T = 256;

  // Symmetric normalization: dinv = rsqrt(1 + in_degree)
  k_deg_init  <<<(N + T - 1) / T, T, 0, stream>>>(dinv, N);
  k_deg_accum <<<(E + T - 1) / T, T, 0, stream>>>(dst, dinv, E);
  k_deg_rsqrt <<<(N + T - 1) / T, T, 0, stream>>>(dinv, N);

  // ---- Layer 1: H = x @ W1 ; AGG = scatter(norm * H) ; AGG = relu(AGG + b1)
  k_gemm_wmma_f32<<<(N + 31) / 32, 32 * (CH / 32), 0, stream>>>(x, W1, Hbuf, N, CIN, CH);
  k_selfloop_init_v4<<<((size_t)N * (CH / 4) + T - 1) / T, T, 0, stream>>>(Hbuf, dinv, b1, AGG, N, CH, 0);
  k_edge_scatter<4><<<((size_t)E * 32 + T - 1) / T, T, 0, stream>>>(src, dst, Hbuf, dinv, AGG, E, CH);
  k_bias_relu_v4<<<((size_t)N * (CH / 4) + T - 1) / T, T, 0, stream>>>(AGG, b1, N, CH);

  // ---- Layer 2: H2 = AGG @ W2 (reuse Hbuf) ; out = scatter(norm * H2) + b2
  k_gemm_wmma_f32<<<(N + 31) / 32, 32 * (CO / 32), 0, stream>>>(AGG, W2, Hbuf, N, CH, CO);
  k_selfloop_init_v4<<<((size_t)N * (CO / 4) + T - 1) / T, T, 0, stream>>>(Hbuf, dinv, b2, out, N, CO, 1);
  k_edge_scatter<2><<<((size_t)E * 32 + T - 1) / T, T, 0, stream>>>(src, dst, Hbuf, dinv, out, E, CO);
}